// ShiftSoftDotAttention_42039139893564
// MI455X (gfx1250) — compile-verified
//
#include <hip/hip_runtime.h>
#include <math.h>

// ---------------------------------------------------------------------------
// Problem constants (ShiftSoftDotAttention)
// ---------------------------------------------------------------------------
#define NB   2048   // batch
#define NS   144    // sequence
#define NQD  512    // query dim
#define NCD  512    // context dim
#define NKT  3      // shift taps
#define NL   48     // NS / 3

typedef float v2f __attribute__((ext_vector_type(2)));
typedef float v8f __attribute__((ext_vector_type(8)));
typedef unsigned int u32x4 __attribute__((ext_vector_type(4)));
typedef unsigned int u32x8 __attribute__((ext_vector_type(8)));

#if defined(__HIP_DEVICE_COMPILE__) && defined(__gfx1250__)
#define CDNA5_DEV 1
#endif

// ---------------------------------------------------------------------------
// fp32 WMMA 16x16x4: D = A(16x4) * B(4x16) + C(16x16)
//   A frag: lane L holds A[M=L&15][K = (L>>4)*2 + {0,1}]   (2 VGPRs)
//   B frag: lane L holds B[K = (L>>4)*2 + {0,1}][N = L&15] (2 VGPRs)
//   C/D  : VGPR r, lanes 0-15 -> M=r, lanes 16-31 -> M=8+r; N = L&15
// ---------------------------------------------------------------------------
#if defined(CDNA5_DEV) && __has_builtin(__builtin_amdgcn_wmma_f32_16x16x4_f32)
__device__ inline v8f wmma4(v2f a, v2f b, v8f c) {
  return __builtin_amdgcn_wmma_f32_16x16x4_f32(
      /*neg_a=*/false, a, /*neg_b=*/false, b,
      /*c_mod=*/(short)0, c, /*reuse_a=*/false, /*reuse_b=*/false);
}
#else
// Host-pass / missing-builtin compile fallback (not numerically meaningful on
// a target without the builtin; real path is the gfx1250 branch above).
__device__ inline v8f wmma4(v2f a, v2f b, v8f c) {
  c[0] += a.x * b.x + a.y * b.y;
  return c;
}
#endif

__device__ inline float wave_sum(float v) {
#pragma unroll
  for (int o = 16; o > 0; o >>= 1) v += __shfl_down(v, o, 32);
  return v;  // valid in lane 0
}

// ---------------------------------------------------------------------------
// Phase A: target[b,c] = sum_q h[b,q] * W_in[c,q]
// GEMM M=2048 N=512 K=512.  Block = 256 threads = 8 waves; each wave owns a
// 16(M) x 64(N) tile with 4 f32 WMMA accumulators.  grid = (8, 16).
// ---------------------------------------------------------------------------
__global__ __launch_bounds__(256) void k_gemm_target(
    const float* __restrict__ H, const float* __restrict__ Win,
    float* __restrict__ Tgt) {
  const int lane = threadIdx.x & 31;
  const int wv   = threadIdx.x >> 5;
  const int ln   = lane & 15;
  const int hi   = lane >> 4;      // 0 or 1
  const int kh   = hi << 1;        // 0 or 2
  const int m0   = (blockIdx.y * 8 + wv) * 16;
  const int n0   = blockIdx.x * 64;

  v8f acc[4] = {};
  const float* hrow = H + (size_t)(m0 + ln) * NQD;

  for (int k0 = 0; k0 < NQD; k0 += 4) {
    const int kb = k0 + kh;
    v2f a; a.x = hrow[kb]; a.y = hrow[kb + 1];
#pragma unroll
    for (int j = 0; j < 4; ++j) {
      const int n = n0 + j * 16 + ln;
      v2f bf; bf.x = Win[(size_t)n * NQD + kb]; bf.y = Win[(size_t)n * NQD + kb + 1];
      acc[j] = wmma4(a, bf, acc[j]);
    }
  }

  const int mrow = m0 + hi * 8;
#pragma unroll
  for (int j = 0; j < 4; ++j) {
    const int n = n0 + j * 16 + ln;
#pragma unroll
    for (int r = 0; r < 8; ++r)
      Tgt[(size_t)(mrow + r) * NCD + n] = acc[j][r];
  }
}

// ---------------------------------------------------------------------------
// Phase B: one workgroup per batch b (256 threads, 8 waves).
//  - TDM loads context[b] (144x512 f32 = 294912 B) into LDS (reads context
//    from HBM exactly once -> minimal traffic at 23.3 TB/s).
//  - logits -> masked softmax -> shift-taps softmax -> circular 3-tap conv
//    -> weighted_context, all out of LDS.
// Dynamic LDS layout (floats): ctx[73728] | tgt[512] | att[144] | shf[144] | red[16]
// ---------------------------------------------------------------------------
__global__ __launch_bounds__(256) void k_attn_shift(
    const float* __restrict__ Ctx, const float* __restrict__ Tgt,
    const int* __restrict__ Mask, const float* __restrict__ H,
    const float* __restrict__ Wsh, const float* __restrict__ Bsh,
    float* __restrict__ AttnOut, float* __restrict__ WC) {
  extern __shared__ float smem[];
  float* ctx = smem;                       // 73728 floats @ LDS offset 0
  float* tgt = smem + NS * NCD;            // 512
  float* att = tgt + NQD;                  // 144
  float* shf = att + NS;                   // 144
  float* red = shf + NS;                   // 16

  const int tid  = threadIdx.x;
  const int lane = tid & 31;
  const int wv   = tid >> 5;
  const int b    = blockIdx.x;
  const float* ctxg = Ctx + (size_t)b * (NS * NCD);

#ifdef CDNA5_DEV
  // --- Tensor Data Mover: async DMA of context[b] -> LDS offset 0 ---------
  if (wv == 0) {
    const unsigned long long ga = (unsigned long long)(const void*)ctxg;
    u32x4 g0;
    g0[0] = 1u;                                           // count=1, user mode
    g0[1] = 0u;                                           // lds_addr = 0
    g0[2] = (unsigned)(ga & 0xffffffffu);                 // global_addr[31:0]
    g0[3] = (unsigned)((ga >> 32) & 0x1ffffffu) | (2u << 30);  // addr[56:32] | type=2
    u32x8 g1;
    g1[0] = 2u << 16;                                     // data_size = 4B
    g1[1] = ((unsigned)NCD & 0xffffu) << 16;              // tensor_dim0 lo16 @ [63:48]
    g1[2] = ((unsigned)NCD >> 16) | (((unsigned)NS & 0xffffu) << 16);  // dim0 hi | dim1 lo
    g1[3] = ((unsigned)NS >> 16) | ((unsigned)NCD << 16); // dim1 hi | tile_dim0
    g1[4] = (unsigned)NS;                                 // tile_dim1 (tile_dim2=0)
    g1[5] = (unsigned)NCD;                                // tensor_dim0_stride[31:0]
    g1[6] = 0u;
    g1[7] = 0u;
    asm volatile("tensor_load_to_lds %0, %1" :: "s"(g0), "s"(g1) : "memory");
  }
#else
  for (int i = tid; i < (NS * NCD) / 4; i += 256)
    ((float4*)ctx)[i] = ((const float4*)ctxg)[i];
#endif

  // --- overlap with DMA: stage target row + compute shift-tap logits ------
  tgt[tid]       = Tgt[(size_t)b * NCD + tid];
  tgt[tid + 256] = Tgt[(size_t)b * NCD + tid + 256];

  if (wv >= 1 && wv <= 3) {
    const int k = wv - 1;
    float acc = 0.f;
    for (int i = lane; i < NQD; i += 32)
      acc += H[(size_t)b * NQD + i] * Wsh[(size_t)k * NQD + i];
    acc = wave_sum(acc);
    if (lane == 0) red[k] = acc + Bsh[k];
  }

#ifdef CDNA5_DEV
  if (wv == 0) __builtin_amdgcn_s_wait_tensorcnt(0);
#endif
  __syncthreads();

  // --- logits[s] = <context[b,s,:], target[b,:]>, masked ------------------
  for (int s = wv; s < NS; s += 8) {
    float acc = 0.f;
    const float* crow = ctx + s * NCD;
    for (int i = lane; i < NCD; i += 32) acc += crow[i] * tgt[i];
    acc = wave_sum(acc);
    if (lane == 0)
      att[s] = Mask[(size_t)b * NS + s] ? -__builtin_inff() : acc;
  }
  __syncthreads();

  // --- softmax over S (redundant per thread; LDS broadcast reads) ---------
  float mx = -__builtin_inff();
  for (int s = 0; s < NS; ++s) mx = fmaxf(mx, att[s]);
  float ssum = 0.f;
  for (int s = 0; s < NS; ++s) ssum += expf(att[s] - mx);
  const float inv = 1.0f / ssum;
  __syncthreads();
  if (tid < NS) {
    const float a = expf(att[tid] - mx) * inv;
    att[tid] = a;
    AttnOut[(size_t)b * NS + tid] = a;
  }
  __syncthreads();

  // --- tap softmax + circular 3-tap shift (per row of 48) -----------------
  const float t0 = red[0], t1 = red[1], t2 = red[2];
  const float tm = fmaxf(t0, fmaxf(t1, t2));
  const float e0 = expf(t0 - tm), e1 = expf(t1 - tm), e2 = expf(t2 - tm);
  const float es = 1.0f / (e0 + e1 + e2);
  const float c0 = e0 * es, c1 = e1 * es, c2 = e2 * es;

  if (tid < NS) {
    const int r = tid / NL, l = tid % NL;
    const int lm = r * NL + (l + NL - 1) % NL;
    const int lp = r * NL + (l + 1) % NL;
    shf[tid] = c0 * att[lm] + c1 * att[tid] + c2 * att[lp];
  }
  __syncthreads();

  // --- weighted_context[d] = sum_s shifted[s] * context[s,d] --------------
  float a0 = 0.f, a1 = 0.f;
  const int d0 = tid, d1 = tid + 256;
  for (int s = 0; s < NS; ++s) {
    const float w = shf[s];
    a0 += w * ctx[s * NCD + d0];
    a1 += w * ctx[s * NCD + d1];
  }
  WC[(size_t)b * NCD + d0] = a0;
  WC[(size_t)b * NCD + d1] = a1;
}

// ---------------------------------------------------------------------------
// Phase C: h_tilde = tanh( [wc, h] @ W_out^T ), M=2048 N=512 K=1024.
// Same WMMA tiling as phase A; K boundary at 512 is 4-aligned so the A-source
// switch is uniform per k-step.
// ---------------------------------------------------------------------------
__global__ __launch_bounds__(256) void k_gemm_out(
    const float* __restrict__ WCm, const float* __restrict__ H,
    const float* __restrict__ Wout, float* __restrict__ Out) {
  const int lane = threadIdx.x & 31;
  const int wv   = threadIdx.x >> 5;
  const int ln   = lane & 15;
  const int hi   = lane >> 4;
  const int kh   = hi << 1;
  const int m0   = (blockIdx.y * 8 + wv) * 16;
  const int n0   = blockIdx.x * 64;
  const int row  = m0 + ln;

  v8f acc[4] = {};
  const float* wcrow = WCm + (size_t)row * NCD;
  const float* hrow  = H   + (size_t)row * NQD;

  for (int k0 = 0; k0 < NQD + NCD; k0 += 4) {
    const int kb = k0 + kh;
    v2f a;
    if (k0 < NCD) { a.x = wcrow[kb];       a.y = wcrow[kb + 1]; }
    else          { a.x = hrow[kb - NCD];  a.y = hrow[kb - NCD + 1]; }
#pragma unroll
    for (int j = 0; j < 4; ++j) {
      const int n = n0 + j * 16 + ln;
      v2f bf;
      bf.x = Wout[(size_t)n * (NQD + NCD) + kb];
      bf.y = Wout[(size_t)n * (NQD + NCD) + kb + 1];
      acc[j] = wmma4(a, bf, acc[j]);
    }
  }

  const int mrow = m0 + hi * 8;
#pragma unroll
  for (int j = 0; j < 4; ++j) {
    const int n = n0 + j * 16 + ln;
#pragma unroll
    for (int r = 0; r < 8; ++r)
      Out[(size_t)(mrow + r) * NQD + n] = tanhf(acc[j][r]);
  }
}

// ---------------------------------------------------------------------------
extern "C" void kernel_launch(void* const* d_in, const int* in_sizes, int n_in,
                              void* d_out, int out_size, void* d_ws, size_t ws_size,
                              hipStream_t stream) {
  const float* h       = (const float*)d_in[0];
  const float* context = (const float*)d_in[1];
  const int*   mask    = (const int*)d_in[2];
  const float* W_in    = (const float*)d_in[3];
  const float* W_shift = (const float*)d_in[4];
  const float* b_shift = (const float*)d_in[5];
  const float* W_out   = (const float*)d_in[6];

  float* h_tilde  = (float*)d_out;                       // [B, QD]
  float* attn_out = (float*)d_out + (size_t)NB * NQD;    // [B, S]

  float* target = (float*)d_ws;                          // [B, CD] scratch
  float* wc     = target + (size_t)NB * NCD;             // [B, CD] scratch

  // Phase A: target = h @ W_in^T  (WMMA f32)
  k_gemm_target<<<dim3(NCD / 64, NB / 128), 256, 0, stream>>>(h, W_in, target);

  // Phase B: attention + shift + weighted context (TDM context -> LDS)
  const size_t ldsBytes = (size_t)(NS * NCD + NQD + NS + NS + 16) * sizeof(float);
  k_attn_shift<<<dim3(NB), 256, ldsBytes, stream>>>(
      context, target, mask, h, W_shift, b_shift, attn_out, wc);

  // Phase C: h_tilde = tanh([wc, h] @ W_out^T)  (WMMA f32 + tanh epilogue)
  k_gemm_out<<<dim3(NQD / 64, NB / 128), 256, 0, stream>>>(wc, h, W_out, h_tilde);
}